// AsymDCT_90941637526307
// MI455X (gfx1250) — compile-verified
//
#include <hip/hip_runtime.h>

// ---------------------------------------------------------------------------
// AsymDCT on MI455X (gfx1250, wave32, WMMA f32 16x16x4)
//
// Math: with T orthonormal (DCT-II), per 8x8 block:
//   x_low  = W x W^T,  W = t^T T[:4,:]          (4x8)
//   x_high = x - Q x Q, Q = T[:4,:]^T T[:4,:]   (8x8 symmetric)
// Pack 2x2 blocks into a 16x16 tile and use block-diagonal 16x16 constants
// Wd, Qd so everything becomes 16x16x16 products = 4 chained
// V_WMMA_F32_16X16X4_F32 per product. Memory-bound: 462 MB total traffic
// -> ~20us floor at 23.3 TB/s; WMMA keeps compute off the critical path.
// ---------------------------------------------------------------------------

typedef __attribute__((ext_vector_type(2))) float v2f;
typedef __attribute__((ext_vector_type(8))) float v8f;

#define IN_SZ   224
#define LOW_SZ  112
#define NT      14                    // 16x16 tiles per image dim
#define LS      20                    // LDS row stride (floats): 80B -> float4-aligned, conflict-light
#define TILES_PER_IMG (NT * NT)       // 196
#define N_IMG   (16 * 64)             // 1024 batch*channels
#define N_TILES (N_IMG * TILES_PER_IMG)
#define WAVES_PER_BLK 8

__device__ __forceinline__ void lds_fence() {
    // CDNA5 split counters: wait for all outstanding LDS ops from this wave.
    asm volatile("s_wait_dscnt 0" ::: "memory");
}

// A-operand chunk c (K columns 4c..4c+3) of a 16x16 matrix in LDS.
// Lane l<16: (M=l, K=4c+{0,1}); lane>=16: (M=l-16, K=4c+{2,3}).
__device__ __forceinline__ v2f ldsA(const float* buf, int l16, int half, int c) {
    int col = 4 * c + 2 * half;
    v2f a;
    a.x = buf[l16 * LS + col];
    a.y = buf[l16 * LS + col + 1];
    return a;
}

// B-operand chunk c (K rows 4c..4c+3) of a 16x16 matrix in LDS.
// Lane l<16: (K=4c+{0,1}, N=l); lane>=16: (K=4c+{2,3}, N=l-16).
__device__ __forceinline__ v2f ldsB(const float* buf, int l16, int half, int c) {
    int row = 4 * c + 2 * half;
    v2f b;
    b.x = buf[row * LS + l16];
    b.y = buf[(row + 1) * LS + l16];
    return b;
}

__global__ __launch_bounds__(256) void asym_dct_kernel(
    const float* __restrict__ x, const float* __restrict__ Tm,
    const float* __restrict__ tm, float* __restrict__ out_low,
    float* __restrict__ out_high)
{
    // [Wd 16xLS][Qd 16xLS][per-wave: X, M1, M2 each 16xLS]
    __shared__ __align__(16) float smem[2 * 16 * LS + WAVES_PER_BLK * 3 * 16 * LS];
    float* wd = smem;
    float* qd = smem + 16 * LS;

    // Build block-diagonal constants from the T (8x8) and t (4x4) inputs.
    {
        int i = threadIdx.x >> 4, j = threadIdx.x & 15;
        float w = 0.f;
        if (i < 4 && j < 8)              { for (int k = 0; k < 4; ++k) w += tm[k*4 + i]       * Tm[k*8 + j];       }
        else if (i >= 4 && i < 8 && j >= 8) { for (int k = 0; k < 4; ++k) w += tm[k*4 + (i-4)] * Tm[k*8 + (j-8)]; }
        wd[i * LS + j] = w;
        float q = 0.f;
        if (i < 8 && j < 8)              { for (int k = 0; k < 4; ++k) q += Tm[k*8 + i]       * Tm[k*8 + j];       }
        else if (i >= 8 && j >= 8)       { for (int k = 0; k < 4; ++k) q += Tm[k*8 + (i-8)]   * Tm[k*8 + (j-8)];   }
        qd[i * LS + j] = q;
    }
    __syncthreads();

    const int wave = threadIdx.x >> 5;
    const int lane = threadIdx.x & 31;
    const int half = lane >> 4;
    const int l16  = lane & 15;

    float* xb = smem + 2 * 16 * LS + wave * 3 * 16 * LS;  // X tile
    float* m1 = xb + 16 * LS;                              // Wd @ X
    float* m2 = m1 + 16 * LS;                              // Qd @ X

    int tile = blockIdx.x * WAVES_PER_BLK + wave;          // grid sized exactly
    int img  = tile / TILES_PER_IMG;
    int rem  = tile - img * TILES_PER_IMG;
    int ti   = rem / NT;
    int tj   = rem - ti * NT;

    const float* xg = x + (size_t)img * IN_SZ * IN_SZ + ti * 16 * IN_SZ + tj * 16;

    // Load 16x16 tile: 64 float4 slots; lane handles slots (lane, lane+32).
    {
        int s0 = lane, s1 = lane + 32;
        int r0 = s0 >> 2, c0 = (s0 & 3) * 4;
        int r1 = s1 >> 2, c1 = (s1 & 3) * 4;
        float4 v0 = *(const float4*)(xg + r0 * IN_SZ + c0);
        float4 v1 = *(const float4*)(xg + r1 * IN_SZ + c1);
        *(float4*)(xb + r0 * LS + c0) = v0;
        *(float4*)(xb + r1 * LS + c1) = v1;
    }
    lds_fence();  // cross-lane RAW: tile stores -> B-layout reads

    // Pass 1: M1 = Wd @ X, M2 = Qd @ X   (K=16 as 4 chunks of K=4)
    v8f accW = {};
    v8f accQ = {};
#pragma unroll
    for (int c = 0; c < 4; ++c) {
        v2f bX = ldsB(xb, l16, half, c);
        v2f aW = ldsA(wd, l16, half, c);
        v2f aQ = ldsA(qd, l16, half, c);
        accW = __builtin_amdgcn_wmma_f32_16x16x4_f32(false, aW, false, bX, (short)0, accW, false, false);
        accQ = __builtin_amdgcn_wmma_f32_16x16x4_f32(false, aQ, false, bX, (short)0, accQ, false, false);
    }

    // Stage C-layout results to LDS (layout transform for the next A-operands).
#pragma unroll
    for (int r = 0; r < 8; ++r) {
        m1[(r + 8 * half) * LS + l16] = accW[r];
        m2[(r + 8 * half) * LS + l16] = accQ[r];
    }
    lds_fence();

    // Pass 2: XL = M1 @ Wd^T (B of a transpose == A-pattern read of Wd),
    //         XR = M2 @ Qd   (Q symmetric).
    v8f accL = {};
    v8f accR = {};
#pragma unroll
    for (int c = 0; c < 4; ++c) {
        v2f aL = ldsA(m1, l16, half, c);
        v2f bW = ldsA(wd, l16, half, c);   // = Wd^T in B layout
        v2f aR = ldsA(m2, l16, half, c);
        v2f bQ = ldsB(qd, l16, half, c);
        accL = __builtin_amdgcn_wmma_f32_16x16x4_f32(false, aL, false, bW, (short)0, accL, false, false);
        accR = __builtin_amdgcn_wmma_f32_16x16x4_f32(false, aR, false, bQ, (short)0, accR, false, false);
    }

    // x_high = X - XR, stored straight from C layout:
    // lane l<16 holds column l rows 0..7; lane>=16 holds column l-16 rows 8..15.
    float* outH = out_high + (size_t)img * IN_SZ * IN_SZ + (ti * 16) * IN_SZ + tj * 16;
#pragma unroll
    for (int r = 0; r < 8; ++r) {
        float xv = xb[(r + 8 * half) * LS + l16];
        outH[(r + 8 * half) * IN_SZ + l16] = xv - accR[r];
    }

    // x_low = XL[0:8, 0:8] -> contiguous 8x8 patch of the 112x112 low image.
    if (half == 0 && l16 < 8) {
        float* outL = out_low + (size_t)img * LOW_SZ * LOW_SZ + (ti * 8) * LOW_SZ + tj * 8;
#pragma unroll
        for (int r = 0; r < 8; ++r) outL[r * LOW_SZ + l16] = accL[r];
    }
}

extern "C" void kernel_launch(void* const* d_in, const int* in_sizes, int n_in,
                              void* d_out, int out_size, void* d_ws, size_t ws_size,
                              hipStream_t stream) {
    const float* x  = (const float*)d_in[0];   // (16,64,224,224) f32
    const float* Tm = (const float*)d_in[1];   // 8x8 DCT
    const float* tm = (const float*)d_in[2];   // 4x4 DCT
    float* out_low  = (float*)d_out;                               // 16*64*112*112
    float* out_high = out_low + (size_t)N_IMG * LOW_SZ * LOW_SZ;   // 16*64*224*224

    dim3 grid(N_TILES / WAVES_PER_BLK);  // 200704 / 8 = 25088, exact
    asym_dct_kernel<<<grid, 256, 0, stream>>>(x, Tm, tm, out_low, out_high);
}